// GCN_57071525430158
// MI455X (gfx1250) — compile-verified
//
#include <hip/hip_runtime.h>
#include <hip/hip_bf16.h>
#include <stdint.h>

typedef __attribute__((ext_vector_type(2))) float v2f;
typedef __attribute__((ext_vector_type(8))) float v8f;

// -------------------- degree / normalization --------------------

__global__ void k_deg_init(float* __restrict__ deg, int n) {
  int i = blockIdx.x * blockDim.x + threadIdx.x;
  if (i < n) deg[i] = 1.0f;  // self-loop contributes 1 at every destination
}

__global__ void k_deg_accum(const long long* __restrict__ dst, float* __restrict__ deg,
                            long long E) {
  long long stride = (long long)gridDim.x * blockDim.x;
  for (long long e = (long long)blockIdx.x * blockDim.x + threadIdx.x; e < E; e += stride) {
    unsafeAtomicAdd(&deg[dst[e]], 1.0f);
  }
}

__global__ void k_deg_rsqrt(float* __restrict__ deg, int n) {
  int i = blockIdx.x * blockDim.x + threadIdx.x;
  if (i < n) deg[i] = rsqrtf(deg[i]);  // deg >= 1 always (self-loops)
}

// -------------------- f32 WMMA GEMM: C[M,N] = A[M,K] @ B[K,N] --------------------
// One wave computes TWO vertically-adjacent 16x16 C tiles (32 rows), sharing the
// B-tile loads. V_WMMA_F32_16X16X4_F32.
// A 16x4 layout: lanes 0-15 hold K = kk+0/kk+1 (vgpr0/vgpr1) for row M=lane;
//                lanes 16-31 hold K = kk+2/kk+3 for row M=lane-16.
// B 4x16 layout mirrors this with N striped across lanes.
// C layout: vgpr j -> row m0+j (lanes 0-15) / m0+j+8 (lanes 16-31), col n0+(lane&15).
__global__ void k_wmma_gemm_f32(const float* __restrict__ A, const float* __restrict__ B,
                                float* __restrict__ C, int M, int N, int K) {
  const int lane = threadIdx.x & 31;
  const int half = lane >> 4;   // 0 or 1
  const int lm   = lane & 15;
  const int m0 = blockIdx.x * 32;        // two 16-row tiles per wave
  const int n0 = blockIdx.y * 16;
  if (m0 >= M) return;
  const bool has2 = (m0 + 16) < M;

  const float* Arow0 = A + (size_t)(m0 + lm) * K;
  const float* Arow1 = has2 ? (Arow0 + (size_t)16 * K) : Arow0;
  const float* Bcol  = B + n0 + lm;

  v8f acc0 = {}, acc1 = {};
  const int Kmain = K & ~3;  // multiple of 4; tail handled separately (no guards here)

#pragma unroll 2
  for (int kk = 0; kk < Kmain; kk += 4) {
    const int ka = kk + 2 * half;
    v2f a0, a1, b;
    a0.x = Arow0[ka];
    a0.y = Arow0[ka + 1];
    a1.x = Arow1[ka];
    a1.y = Arow1[ka + 1];
    b.x  = Bcol[(size_t)ka * N];
    b.y  = Bcol[(size_t)(ka + 1) * N];
    // 8 args: (neg_a, A, neg_b, B, c_mod, C, reuse_a, reuse_b)
    acc0 = __builtin_amdgcn_wmma_f32_16x16x4_f32(false, a0, false, b, (short)0, acc0,
                                                 false, false);
    acc1 = __builtin_amdgcn_wmma_f32_16x16x4_f32(false, a1, false, b, (short)0, acc1,
                                                 false, false);
  }

  if (Kmain < K) {  // single guarded tail step covering K%4 leftover columns
    const int ka = Kmain + 2 * half;
    v2f a0, a1, b;
    a0.x = (ka     < K) ? Arow0[ka]     : 0.0f;
    a0.y = (ka + 1 < K) ? Arow0[ka + 1] : 0.0f;
    a1.x = (ka     < K) ? Arow1[ka]     : 0.0f;
    a1.y = (ka + 1 < K) ? Arow1[ka + 1] : 0.0f;
    b.x  = (ka     < K) ? Bcol[(size_t)ka * N]       : 0.0f;
    b.y  = (ka + 1 < K) ? Bcol[(size_t)(ka + 1) * N] : 0.0f;
    acc0 = __builtin_amdgcn_wmma_f32_16x16x4_f32(false, a0, false, b, (short)0, acc0,
                                                 false, false);
    acc1 = __builtin_amdgcn_wmma_f32_16x16x4_f32(false, a1, false, b, (short)0, acc1,
                                                 false, false);
  }

  float* Cp0 = C + (size_t)(m0 + half * 8) * N + n0 + lm;
#pragma unroll
  for (int j = 0; j < 8; ++j) Cp0[(size_t)j * N] = acc0[j];
  if (has2) {
    float* Cp1 = Cp0 + (size_t)16 * N;
#pragma unroll
    for (int j = 0; j < 8; ++j) Cp1[(size_t)j * N] = acc1[j];
  }
}

// -------------------- tiny GEMM for layer 3 (K=32, N=2) --------------------

__global__ void k_gemm_n2(const float* __restrict__ h, const float* __restrict__ W,
                          float* __restrict__ out, int n, int K) {
  int i = blockIdx.x * blockDim.x + threadIdx.x;
  if (i >= n) return;
  const float* hp = h + (size_t)i * K;
  float z0 = 0.0f, z1 = 0.0f;
#pragma unroll 8
  for (int k = 0; k < K; ++k) {
    float v = hp[k];
    z0 += v * W[2 * k + 0];
    z1 += v * W[2 * k + 1];
  }
  out[2 * (size_t)i + 0] = z0;
  out[2 * (size_t)i + 1] = z1;
}

// -------------------- GCN aggregation pieces --------------------

// A[i][c] = dinv[i]^2 * H[i][c]   (self-loop message; norm = dinv*dinv)
template <int LOGF>
__global__ void k_selfloop_init(const float* __restrict__ H, const float* __restrict__ dinv,
                                float* __restrict__ A, long long total) {
  long long i = (long long)blockIdx.x * blockDim.x + threadIdx.x;
  if (i >= total) return;
  long long node = i >> LOGF;
  float di = dinv[node];
  A[i] = di * di * H[i];
}

// A[dst][c] += dinv[src]*dinv[dst] * H[src][c]   over all edges, edge-parallel
template <int LOGF>
__global__ void k_edge_agg(const long long* __restrict__ src, const long long* __restrict__ dst,
                           const float* __restrict__ dinv, const float* __restrict__ H,
                           float* __restrict__ A, long long E) {
  const int F = 1 << LOGF;
  const long long total = E << LOGF;
  const long long stride = (long long)gridDim.x * blockDim.x;
  for (long long i = (long long)blockIdx.x * blockDim.x + threadIdx.x; i < total; i += stride) {
    long long e = i >> LOGF;
    int c = (int)(i & (long long)(F - 1));
    long long s = src[e];
    long long d = dst[e];
    float nrm = dinv[s] * dinv[d];
    unsafeAtomicAdd(&A[d * F + c], nrm * H[s * F + c]);
  }
}

// A[i][c] = relu(A[i][c] + b[c])
template <int LOGF>
__global__ void k_bias_relu(float* __restrict__ A, const float* __restrict__ b,
                            long long total) {
  const int F = 1 << LOGF;
  long long i = (long long)blockIdx.x * blockDim.x + threadIdx.x;
  if (i >= total) return;
  int c = (int)(i & (long long)(F - 1));
  A[i] = fmaxf(A[i] + b[c], 0.0f);
}

// out[i][:] = log_softmax(out[i][:] + b3), 2 classes, stable
__global__ void k_finalize(float* __restrict__ out, const float* __restrict__ b3, int n) {
  int i = blockIdx.x * blockDim.x + threadIdx.x;
  if (i >= n) return;
  float z0 = out[2 * (size_t)i + 0] + b3[0];
  float z1 = out[2 * (size_t)i + 1] + b3[1];
  float m = fmaxf(z0, z1);
  float lse = m + logf(expf(z0 - m) + expf(z1 - m));
  out[2 * (size_t)i + 0] = z0 - lse;
  out[2 * (size_t)i + 1] = z1 - lse;
}

// -------------------- launch --------------------

extern "C" void kernel_launch(void* const* d_in, const int* in_sizes, int n_in,
                              void* d_out, int out_size, void* d_ws, size_t ws_size,
                              hipStream_t stream) {
  const float* x  = (const float*)d_in[0];
  const long long* edge_index = (const long long*)d_in[1];  // int64 in reference
  const float* W1 = (const float*)d_in[2];
  const float* b1 = (const float*)d_in[3];
  const float* W2 = (const float*)d_in[4];
  const float* b2 = (const float*)d_in[5];
  const float* W3 = (const float*)d_in[6];
  const float* b3 = (const float*)d_in[7];

  const int K1 = 489, F1 = 64, F2 = 32;
  const int n = in_sizes[0] / K1;                  // 100000
  const long long E = (long long)in_sizes[1] / 2;  // 3200000
  const long long* srcp = edge_index;
  const long long* dstp = edge_index + E;

  // workspace layout (floats)
  float* ws   = (float*)d_ws;
  float* dinv = ws;                              // n (degree then rsqrt in place)
  float* H1   = dinv + n;                        // n*64
  float* A1   = H1 + (size_t)n * F1;             // n*64 (becomes h1 after relu)
  float* H2   = A1 + (size_t)n * F1;             // n*32
  float* A2   = H2 + (size_t)n * F2;             // n*32 (becomes h2 after relu)
  float* H3   = A2 + (size_t)n * F2;             // n*2
  float* A3   = (float*)d_out;                   // n*2, finalized in place

  const int TB = 256;
  const int edgeBlocks = 8192;  // grid-stride over edge work
  dim3 tb(TB);

  // degrees + symmetric normalization
  k_deg_init<<<dim3((n + TB - 1) / TB), tb, 0, stream>>>(dinv, n);
  k_deg_accum<<<dim3(edgeBlocks), tb, 0, stream>>>(dstp, dinv, E);
  k_deg_rsqrt<<<dim3((n + TB - 1) / TB), tb, 0, stream>>>(dinv, n);

  // layer 1: H1 = x @ W1 (WMMA f32); agg; +b1; relu
  k_wmma_gemm_f32<<<dim3((n + 31) / 32, F1 / 16), dim3(32), 0, stream>>>(x, W1, H1, n, F1, K1);
  k_selfloop_init<6><<<dim3((int)(((long long)n * F1 + TB - 1) / TB)), tb, 0, stream>>>(
      H1, dinv, A1, (long long)n * F1);
  k_edge_agg<6><<<dim3(edgeBlocks), tb, 0, stream>>>(srcp, dstp, dinv, H1, A1, E);
  k_bias_relu<6><<<dim3((int)(((long long)n * F1 + TB - 1) / TB)), tb, 0, stream>>>(
      A1, b1, (long long)n * F1);

  // layer 2: H2 = h1 @ W2 (WMMA f32); agg; +b2; relu
  k_wmma_gemm_f32<<<dim3((n + 31) / 32, F2 / 16), dim3(32), 0, stream>>>(A1, W2, H2, n, F2, F1);
  k_selfloop_init<5><<<dim3((int)(((long long)n * F2 + TB - 1) / TB)), tb, 0, stream>>>(
      H2, dinv, A2, (long long)n * F2);
  k_edge_agg<5><<<dim3(edgeBlocks), tb, 0, stream>>>(srcp, dstp, dinv, H2, A2, E);
  k_bias_relu<5><<<dim3((int)(((long long)n * F2 + TB - 1) / TB)), tb, 0, stream>>>(
      A2, b2, (long long)n * F2);

  // layer 3: H3 = h2 @ W3 (N=2, scalar); agg into d_out; +b3; log_softmax
  k_gemm_n2<<<dim3((n + TB - 1) / TB), tb, 0, stream>>>(A2, W3, H3, n, F2);
  k_selfloop_init<1><<<dim3((int)(((long long)n * 2 + TB - 1) / TB)), tb, 0, stream>>>(
      H3, dinv, A3, (long long)n * 2);
  k_edge_agg<1><<<dim3(edgeBlocks), tb, 0, stream>>>(srcp, dstp, dinv, H3, A3, E);
  k_finalize<<<dim3((n + TB - 1) / TB), tb, 0, stream>>>(A3, b3, n);
}